// DAGKT_12764642803827
// MI455X (gfx1250) — compile-verified
//
#include <hip/hip_runtime.h>
#include <hip/hip_bf16.h>

#define NN 131072
#define DD 64
#define RR 3
#define NBASES 2
#define BSUB 4096

typedef float v2f __attribute__((ext_vector_type(2)));
typedef float v8f __attribute__((ext_vector_type(8)));

// ---------------------------------------------------------------------------
// Weight prep (per conv): build 4 combined 64x64 matrices
//   mt=0..2 : W_r = C[r,0]*V[0] + C[r,1]*V[1]
//   mt=3    : self-loop W
// stored k-pair interleaved: Wc2[((mt*32 + k/2)*64 + n)*2 + (k&1)]
// so a lane's (W[ka][n], W[ka+1][n]) is one aligned b64 load.
// ---------------------------------------------------------------------------
__global__ __launch_bounds__(256)
void prep_weights(const float* __restrict__ V,  // [NB,64,64]
                  const float* __restrict__ C,  // [R,NB]
                  const float* __restrict__ W,  // [64,64]
                  float* __restrict__ Wc2)      // [4*64*64]
{
    const int i = blockIdx.x * blockDim.x + threadIdx.x;   // 0..16383
    const int mt  = i >> 12;
    const int idx = i & 4095;                              // k*64 + n
    const int k = idx >> 6;
    const int n = idx & 63;
    float w;
    if (mt < 3) w = C[mt * NBASES + 0] * V[idx] + C[mt * NBASES + 1] * V[4096 + idx];
    else        w = W[idx];
    Wc2[((mt * 32 + (k >> 1)) * 64 + n) * 2 + (k & 1)] = w;
}

// ---------------------------------------------------------------------------
// Fused GEMM: hR[r] = h @ W_r (r=0..2), agg = h @ W_self + bias.
// grid.x = N/16 row blocks; block = 128 = 4 waves; wave mt owns matrix mt.
// Branch-free inner loop: 1 A b64 + 4 B b64 + 4 f32 WMMAs per k-step.
// ---------------------------------------------------------------------------
__global__ __launch_bounds__(128)
void relconv_gemm(const float* __restrict__ h,     // [N,64]
                  const float* __restrict__ Wc2,   // [4][32][64][2]
                  const float* __restrict__ bias,  // [64]
                  float* __restrict__ hR,          // [R,N,64]
                  float* __restrict__ agg)         // [N,64]
{
    const int lane = threadIdx.x & 31;
    const int mt   = threadIdx.x >> 5;        // wave id 0..3 (uniform per wave)
    const int row0 = blockIdx.x * 16;
    const int m    = lane & 15;
    const int kh   = lane >> 4;               // K-half: 0 -> K=0,1 ; 1 -> K=2,3

    v8f acc[4] = {};

    const float* hrow  = h + (size_t)(row0 + m) * DD;
    const float* wbase = Wc2 + (size_t)mt * 4096;

    for (int kk = 0; kk < DD; kk += 4) {
        const int ka = kk + 2 * kh;           // even
        const v2f a = *(const v2f*)(hrow + ka);
#pragma unroll
        for (int t = 0; t < 4; ++t) {
            const int n = t * 16 + m;
            const v2f b = *(const v2f*)(wbase + (((ka >> 1) * 64) + n) * 2);
            acc[t] = __builtin_amdgcn_wmma_f32_16x16x4_f32(
                false, a, false, b, (short)0, acc[t], false, false);
        }
    }

    float* outp = (mt < 3) ? (hR + ((size_t)mt * NN + row0) * DD)
                           : (agg + (size_t)row0 * DD);
    const float bsc = (mt == 3) ? 1.f : 0.f;  // branchless bias select
#pragma unroll
    for (int t = 0; t < 4; ++t) {
        const int n = t * 16 + m;
        const float bv = bias[n] * bsc;
#pragma unroll
        for (int j = 0; j < 8; ++j) {
            // C/D layout: lane<16 -> M=j ; lane>=16 -> M=8+j
            outp[(size_t)(j + 8 * kh) * DD + n] = acc[t][j] + bv;
        }
    }
}

// ---------------------------------------------------------------------------
// Edge scatter: one wave32 per edge.
//   agg[dst] += norm[e] * hR[etype[e]][src[e]]
// Gather via b64 (lane owns components 2*lane, 2*lane+1); L2-resident atomics.
// ---------------------------------------------------------------------------
__global__ __launch_bounds__(256)
void edge_scatter(const int* __restrict__ src, const int* __restrict__ dst,
                  const int* __restrict__ etype, const float* __restrict__ norm,
                  const float* __restrict__ hR, float* __restrict__ agg, int nedges)
{
    const int wave = (int)((blockIdx.x * blockDim.x + threadIdx.x) >> 5);
    const int lane = threadIdx.x & 31;
    if (wave >= nedges) return;

    const int   s = src[wave];
    const int   d = dst[wave];
    const int   r = etype[wave];
    const float w = norm[wave];

    const float* sp = hR + ((size_t)r * NN + (size_t)s) * DD;
    float*       dp = agg + (size_t)d * DD;

    const v2f v = *(const v2f*)(sp + lane * 2);
    atomicAdd(dp + lane * 2 + 0, v.x * w);
    atomicAdd(dp + lane * 2 + 1, v.y * w);
}

// ---------------------------------------------------------------------------
// Activations (in place on agg). Leaky variant also writes the concat buffer
// for the first BSUB rows (column block `layer`).
// ---------------------------------------------------------------------------
__global__ __launch_bounds__(256)
void act_elu(float* __restrict__ a, int n)
{
    const int i = blockIdx.x * blockDim.x + threadIdx.x;
    if (i >= n) return;
    const float x = a[i];
    a[i] = (x > 0.f) ? x : (__expf(x) - 1.f);
}

__global__ __launch_bounds__(256)
void act_leaky_cat(float* __restrict__ a, float* __restrict__ cat, int layer, int n)
{
    const int i = blockIdx.x * blockDim.x + threadIdx.x;
    if (i >= n) return;
    const float x = a[i];
    const float y = (x > 0.f) ? x : 0.01f * x;
    a[i] = y;
    const int row = i >> 6;
    const int col = i & 63;
    if (row < BSUB) cat[(size_t)row * (3 * DD) + layer * DD + col] = y;
}

// ---------------------------------------------------------------------------
// MLP head: per-row block of 128 threads; thread t computes hid[t], LDS
// reduction for the single output, sigmoid.
// ---------------------------------------------------------------------------
__global__ __launch_bounds__(128)
void mlp_head(const float* __restrict__ cat, const float* __restrict__ w1,
              const float* __restrict__ b1, const float* __restrict__ w2,
              const float* __restrict__ b2, float* __restrict__ out)
{
    __shared__ float sh[128];
    const int row = blockIdx.x;
    const int t   = threadIdx.x;

    const float* c = cat + (size_t)row * (3 * DD);
    const float* w = w1 + (size_t)t * (3 * DD);
    float s = b1[t];
#pragma unroll 8
    for (int k = 0; k < 3 * DD; ++k) s += c[k] * w[k];
    s = fmaxf(s, 0.f);
    sh[t] = s * w2[t];
    __syncthreads();
    for (int off = 64; off > 0; off >>= 1) {
        if (t < off) sh[t] += sh[t + off];
        __syncthreads();
    }
    if (t == 0) out[row] = 1.f / (1.f + __expf(-(sh[0] + b2[0])));
}

// ---------------------------------------------------------------------------
extern "C" void kernel_launch(void* const* d_in, const int* in_sizes, int n_in,
                              void* d_out, int out_size, void* d_ws, size_t ws_size,
                              hipStream_t stream)
{
    const float* x     = (const float*)d_in[0];
    const int*   src   = (const int*)d_in[1];
    const int*   dst   = (const int*)d_in[2];
    const int*   etype = (const int*)d_in[3];
    const float* mask  = (const float*)d_in[4];
    const float* mask2 = (const float*)d_in[5];
    const float* lV    = (const float*)d_in[6];
    const float* lC    = (const float*)d_in[7];
    const float* lW    = (const float*)d_in[8];
    const float* lB    = (const float*)d_in[9];
    const float* gV    = (const float*)d_in[10];
    const float* gC    = (const float*)d_in[11];
    const float* gW    = (const float*)d_in[12];
    const float* gB    = (const float*)d_in[13];
    const float* w1    = (const float*)d_in[14];
    const float* b1    = (const float*)d_in[15];
    const float* w2    = (const float*)d_in[16];
    const float* b2    = (const float*)d_in[17];

    const int nedges = in_sizes[1];

    // workspace carve-up (floats):
    //   hR   : R*N*D = 25,165,824     aggA/aggB : N*D each
    //   cat  : B*3D  = 786,432        Wc2 : 16,384       total ~171 MB
    float* ws   = (float*)d_ws;
    float* hR   = ws;
    float* aggA = hR + (size_t)RR * NN * DD;
    float* aggB = aggA + (size_t)NN * DD;
    float* cat  = aggB + (size_t)NN * DD;
    float* Wc2  = cat + (size_t)BSUB * 3 * DD;

    const int n_elem      = NN * DD;
    const int act_blocks  = (n_elem + 255) / 256;
    const int edge_blocks = (nedges + 7) / 8;   // 8 waves per 256-thread block

    const float* hcur = x;
    for (int k = 0; k < 6; ++k) {
        const int  layer    = k >> 1;
        const bool isGlobal = (k & 1) != 0;
        float* agg = ((k & 1) == 0) ? aggA : aggB;

        const float* V    = (isGlobal ? gV : lV) + (size_t)layer * NBASES * DD * DD;
        const float* C    = (isGlobal ? gC : lC) + (size_t)layer * RR * NBASES;
        const float* W    = (isGlobal ? gW : lW) + (size_t)layer * DD * DD;
        const float* bias = (isGlobal ? gB : lB) + (size_t)layer * DD;
        const float* nrm  = isGlobal ? mask2 : mask;

        prep_weights<<<64, 256, 0, stream>>>(V, C, W, Wc2);
        relconv_gemm<<<NN / 16, 128, 0, stream>>>(hcur, Wc2, bias, hR, agg);
        edge_scatter<<<edge_blocks, 256, 0, stream>>>(src, dst, etype, nrm, hR, agg, nedges);
        if (!isGlobal)
            act_elu<<<act_blocks, 256, 0, stream>>>(agg, n_elem);
        else
            act_leaky_cat<<<act_blocks, 256, 0, stream>>>(agg, cat, layer, n_elem);
        hcur = agg;
    }

    mlp_head<<<BSUB, 128, 0, stream>>>(cat, w1, b1, w2, b2, (float*)d_out);
}